// RBF_random_16527034155476
// MI455X (gfx1250) — compile-verified
//
#include <hip/hip_runtime.h>

typedef __attribute__((ext_vector_type(2))) float v2f;
typedef __attribute__((ext_vector_type(4))) float v4f;
typedef __attribute__((ext_vector_type(8))) float v8f;

#define B_ROWS 131072
#define OUT_N  512
#define WAVES_PER_BLOCK 8

// Full distance algebra inside the WMMA:
//   A row m  = (x0, x1, x2, 1)             (16x4)
//   B col n  = (-2c0, -2c1, -2c2, |c|^2)   (4x16)
//   C[r]     = |x_m|^2
//   => D = A*B + C = |x|^2 - 2 x.c + |c|^2 = d^2  directly.
// Epilogue: d = sqrt(|d2|)  (fabs = free src modifier, replaces clamp);
//           out = (1-d)^6 * fma(d, fma(d, 35w/3, 6w), w)
__global__ __launch_bounds__(256) void rbf_wmma_kernel(
    const float* __restrict__ x,
    const float* __restrict__ centers,
    const float* __restrict__ weights,
    float* __restrict__ out)
{
    __shared__ float bpack[OUT_N][4];  // {-2c0, -2c1, -2c2, |c|^2}
    __shared__ float cwco[OUT_N][4];   // {35w/3, 6w, w, 0}  (16B aligned)

    for (int i = threadIdx.x; i < OUT_N; i += 256) {
        float a = centers[i * 3 + 0];
        float b = centers[i * 3 + 1];
        float c = centers[i * 3 + 2];
        float w = weights[i];
        bpack[i][0] = -2.0f * a;
        bpack[i][1] = -2.0f * b;
        bpack[i][2] = -2.0f * c;
        bpack[i][3] = a * a + b * b + c * c;
        cwco[i][0] = w * (35.0f / 3.0f);
        cwco[i][1] = w * 6.0f;
        cwco[i][2] = w;
        cwco[i][3] = 0.0f;
    }
    __syncthreads();

    const int lane = threadIdx.x & 31;
    const int wave = threadIdx.x >> 5;
    const int m16  = lane & 15;
    const int hi   = lane >> 4;            // 0: lanes 0-15 (K0,K1), 1: lanes 16-31 (K2,K3)

    const int strip = blockIdx.x * WAVES_PER_BLOCK + wave;
    const int row0  = strip * 16;

    // ---- A fragment: 16 rows of x, K3 = 1.0 (carries |c|^2 from B) ----
    const float* xr = x + (size_t)(row0 + m16) * 3;
    float x0 = xr[0], x1 = xr[1], x2 = xr[2];
    v2f afrag;
    afrag.x = hi ? x2   : x0;
    afrag.y = hi ? 1.0f : x1;

    // |x|^2 of row m16; gather into C-operand layout (VGPR r -> M = r + 8*hi).
    float xnorm = x0 * x0 + x1 * x1 + x2 * x2;
    v8f accin;
#pragma unroll
    for (int r = 0; r < 8; ++r)
        accin[r] = __shfl(xnorm, r + 8 * hi, 32);

    // Output base: element (row0 + 8*hi, m16); r advances M by 1, tile advances N by 16.
    float* po = out + (size_t)(row0 + 8 * hi) * OUT_N + m16;

    const v2f* bp = (const v2f*)&bpack[0][0];   // index: col*2 + hi

#pragma unroll 4
    for (int t = 0; t < OUT_N / 16; ++t) {
        const int col = t * 16 + m16;

        // ---- B fragment: one ds_load_b64 per lane ----
        v2f bfrag = bp[col * 2 + hi];
        // polynomial coefficients for this column (ds_load_b96)
        v4f co = *(const v4f*)&cwco[col][0];

        v8f acc = __builtin_amdgcn_wmma_f32_16x16x4_f32(
            /*neg_a=*/false, afrag, /*neg_b=*/false, bfrag,
            /*c_mod=*/(short)0, accin, /*reuse_a=*/false, /*reuse_b=*/false);

#pragma unroll
        for (int r = 0; r < 8; ++r) {
            // |.| folds into v_sqrt_f32 source modifier; d2<0 only by rounding
            float d  = __builtin_amdgcn_sqrtf(__builtin_fabsf(acc[r]));
            float u  = 1.0f - d;
            float u2 = u * u;
            float u4 = u2 * u2;
            float u6 = u4 * u2;
            float q  = __builtin_fmaf(d, __builtin_fmaf(d, co.x, co.y), co.z);
            po[r * OUT_N + t * 16] = u6 * q;
        }
    }
}

extern "C" void kernel_launch(void* const* d_in, const int* in_sizes, int n_in,
                              void* d_out, int out_size, void* d_ws, size_t ws_size,
                              hipStream_t stream) {
    const float* x       = (const float*)d_in[0];   // (131072, 3)
    const float* centers = (const float*)d_in[1];   // (512, 3)
    const float* weights = (const float*)d_in[2];   // (512,)
    float* out           = (float*)d_out;           // (131072, 512)

    const int strips = B_ROWS / 16;                       // 8192
    const int blocks = strips / WAVES_PER_BLOCK;          // 1024
    rbf_wmma_kernel<<<blocks, 256, 0, stream>>>(x, centers, weights, out);
}